// WordSpeechContinuousFusion_4896262718144
// MI455X (gfx1250) — compile-verified
//
#include <hip/hip_runtime.h>
#include <hip/hip_bf16.h>

// Problem constants (match reference)
#define BB 16
#define SS 2048
#define DD 512
#define FUSE_THR 0.5f

typedef __attribute__((ext_vector_type(2))) float v2f;
typedef __attribute__((ext_vector_type(8))) float v8f;

// ---------------------------------------------------------------------------
// K1: per-frame dot products  u[f] = x[f].w_score[:D], v[f] = x[f].w_score[D:]
// One wave32 per frame; coalesced strided loads; shuffle reduction.
// ---------------------------------------------------------------------------
__global__ __launch_bounds__(256) void k1_scores(const float* __restrict__ x,
                                                 const float* __restrict__ wscore,
                                                 float* __restrict__ u,
                                                 float* __restrict__ v) {
    const int wid  = (blockIdx.x * 256 + threadIdx.x) >> 5;   // frame id in [0, B*S)
    const int lane = threadIdx.x & 31;
    const float* row = x + (size_t)wid * DD;
    float su = 0.0f, sv = 0.0f;
    #pragma unroll 4
    for (int j = lane; j < DD; j += 32) {
        const float xv = row[j];
        su += xv * wscore[j];
        sv += xv * wscore[DD + j];
    }
    #pragma unroll
    for (int off = 16; off > 0; off >>= 1) {
        su += __shfl_xor(su, off, 32);
        sv += __shfl_xor(sv, off, 32);
    }
    if (lane == 0) { u[wid] = su; v[wid] = sv; }
}

// ---------------------------------------------------------------------------
// K2: per-batch segment-id scan. One block of 256 threads per batch, each
// thread owns 8 consecutive frames. Produces segstart[] (init to SS) and
// n_words (to workspace and to the int32 tail of d_out).
// ---------------------------------------------------------------------------
__global__ __launch_bounds__(256) void k2_scan(const float* __restrict__ u,
                                               const float* __restrict__ v,
                                               const float* __restrict__ bscore,
                                               int* __restrict__ segstart,
                                               int* __restrict__ nwords,
                                               int* __restrict__ out_nwords) {
    const int b   = blockIdx.x;
    const int tid = threadIdx.x;
    const float bsc = bscore[0];

    // init segstart for this batch to SS (so end-of-segment lookup is uniform)
    #pragma unroll
    for (int i = 0; i < 8; ++i) segstart[b * SS + tid * 8 + i] = SS;

    __shared__ int sm[256];

    int stv[8];
    int inc[8];
    int sum = 0;
    #pragma unroll
    for (int i = 0; i < 8; ++i) {
        const int t = tid * 8 + i;
        int st = 0;
        if (t > 0) {
            const float score = u[b * SS + t - 1] + v[b * SS + t] + bsc;
            st = (score > FUSE_THR) ? 0 : 1;   // new segment when NOT fused
        }
        stv[i] = st;
        sum += st;
        inc[i] = sum;                           // inclusive within-thread scan
    }
    sm[tid] = sum;
    __syncthreads();
    // Hillis-Steele inclusive scan over the 256 per-thread sums
    for (int off = 1; off < 256; off <<= 1) {
        int add = (tid >= off) ? sm[tid - off] : 0;
        __syncthreads();
        sm[tid] += add;
        __syncthreads();
    }
    const int prefix = (tid > 0) ? sm[tid - 1] : 0;

    #pragma unroll
    for (int i = 0; i < 8; ++i) {
        const int t   = tid * 8 + i;
        const int seg = prefix + inc[i];
        if (t == 0 || stv[i]) segstart[b * SS + seg] = t;   // segment start frame
        if (t == SS - 1) {
            const int nw = seg + 1;
            nwords[b] = nw;
            out_nwords[b] = nw;                 // int32 bits into output tail
        }
    }
}

// ---------------------------------------------------------------------------
// K3: per-segment mean. One wave32 per (batch, word-slot); streams the
// contiguous run with float4 loads (128 contiguous floats per wave per pass).
// Rows with k >= n_words are left untouched (masked at K4 store).
// ---------------------------------------------------------------------------
__global__ __launch_bounds__(256) void k3_segmean(const float* __restrict__ x,
                                                  const int* __restrict__ segstart,
                                                  const int* __restrict__ nwords,
                                                  float* __restrict__ seg_mean) {
    const int wid  = (blockIdx.x * 256 + threadIdx.x) >> 5;  // (b,k) pair
    const int lane = threadIdx.x & 31;
    const int b = wid >> 11;          // /SS
    const int k = wid & (SS - 1);
    if (k >= nwords[b]) return;

    const int start = segstart[b * SS + k];
    const int end   = (k == SS - 1) ? SS : segstart[b * SS + k + 1];

    float4 acc[4];
    #pragma unroll
    for (int p = 0; p < 4; ++p) { acc[p].x = acc[p].y = acc[p].z = acc[p].w = 0.0f; }

    for (int t = start; t < end; ++t) {
        const float4* row = (const float4*)(x + (size_t)(b * SS + t) * DD);
        #pragma unroll
        for (int p = 0; p < 4; ++p) {
            const float4 r = row[p * 32 + lane];
            acc[p].x += r.x; acc[p].y += r.y; acc[p].z += r.z; acc[p].w += r.w;
        }
    }
    const float inv = 1.0f / (float)(end - start);
    float4* dst = (float4*)(seg_mean + (size_t)(b * SS + k) * DD);
    #pragma unroll
    for (int p = 0; p < 4; ++p) {
        float4 r;
        r.x = acc[p].x * inv; r.y = acc[p].y * inv;
        r.z = acc[p].z * inv; r.w = acc[p].w * inv;
        dst[p * 32 + lane] = r;
    }
}

// ---------------------------------------------------------------------------
// K4: words = seg_mean @ w_comb + b_comb, masked by k < n_words[b].
// LDS-staged, double-buffered f32 WMMA GEMM (V_WMMA_F32_16X16X4_F32).
//
// Block = 256 threads (8 waves) computes a 128(M) x 128(N) tile.
// Wave computes 16(M) x 128(N) with 8 v8f accumulators.
// K streamed in 16-wide chunks through LDS (register-prefetch double buffer,
// one barrier per chunk).
//
// LDS layouts (fragment-native, one ds_load_b64 per fragment):
//   W: ldsW[buf][kp*288 + n*2 + par]  (kp = K-pair 0..7, n = 0..127, par = K&1)
//      row stride 288 dwords -> the two lane-halves hit banks 0..31 / 32..63.
//   A: ldsA[buf][wave*256 + kp*32 + m*2 + par]  (kp stride 32 -> halves
//      naturally conflict-free).
//
// Fragment mapping (16x16x4 f32):
//   A frag: lane L (m=L%16, h=L/16): K = k0 + 2h + {0,1}
//   B frag: lane L (n=L%16, h=L/16): K = k0 + 2h + {0,1}
//   C/D:    lane L: N = L%16; VGPR r: M = r + 8h
// ---------------------------------------------------------------------------
__global__ __launch_bounds__(256) void k4_gemm(const float* __restrict__ A,     // seg_mean [B*S, D]
                                               const float* __restrict__ W,     // w_comb  [D, D]
                                               const float* __restrict__ bias,  // b_comb  [D]
                                               const int* __restrict__ nwords,
                                               float* __restrict__ out) {
    __shared__ float ldsW[2][8 * 288];      // 2 x 9216 B
    __shared__ float ldsA[2][8 * 256];      // 2 x 8192 B

    const int tid     = threadIdx.x;
    const int lane    = tid & 31;
    const int wave    = tid >> 5;
    const int m_local = lane & 15;
    const int half    = lane >> 4;

    const int row_tile = blockIdx.x * 8 + wave;     // M/16 tile index (0..2047)
    const int n0       = blockIdx.y * 128;          // N group base

    // --- staging thread mappings -------------------------------------------
    // W chunk: 16 K-rows x 128 N. Thread -> K-pair kpw (0..7), 4 cols at nqw.
    const int kpw = tid >> 5;
    const int nqw = (tid & 31) << 2;
    const float* wsrc = W + (size_t)(2 * kpw) * DD + n0 + nqw;
    // A chunk: 128 M-rows x 16 K. Thread -> row ra (0..127), 8 cols at hca.
    const int ra  = tid >> 1;
    const int hca = (tid & 1) << 3;
    const float* asrc = A + (size_t)(blockIdx.x * 128 + ra) * DD + hca;

    v8f acc[8] = {};
    float4 wg0, wg1, ag0, ag1;

    // preload chunk 0 into registers
    {
        const float* wp = wsrc;                  // kc = 0
        wg0 = *(const float4*)(wp);
        wg1 = *(const float4*)(wp + DD);
        ag0 = *(const float4*)(asrc);
        ag1 = *(const float4*)(asrc + 4);
    }
    // store chunk 0 into LDS buf 0
    {
        float* wd = &ldsW[0][kpw * 288 + nqw * 2];
        float2 t;
        t.x = wg0.x; t.y = wg1.x; *(float2*)(wd + 0) = t;
        t.x = wg0.y; t.y = wg1.y; *(float2*)(wd + 2) = t;
        t.x = wg0.z; t.y = wg1.z; *(float2*)(wd + 4) = t;
        t.x = wg0.w; t.y = wg1.w; *(float2*)(wd + 6) = t;
        float* ad = &ldsA[0][(ra >> 4) * 256 + (hca >> 1) * 32 + (ra & 15) * 2];
        t.x = ag0.x; t.y = ag0.y; *(float2*)(ad + 0 * 32) = t;
        t.x = ag0.z; t.y = ag0.w; *(float2*)(ad + 1 * 32) = t;
        t.x = ag1.x; t.y = ag1.y; *(float2*)(ad + 2 * 32) = t;
        t.x = ag1.z; t.y = ag1.w; *(float2*)(ad + 3 * 32) = t;
    }
    __syncthreads();

    const int NC = DD / 16;                      // 32 K-chunks
    for (int c = 0; c < NC; ++c) {
        const int buf = c & 1;
        // prefetch next chunk into registers (uniform branch)
        if (c + 1 < NC) {
            const int kc = (c + 1) * 16;
            const float* wp = wsrc + (size_t)kc * DD;
            wg0 = *(const float4*)(wp);
            wg1 = *(const float4*)(wp + DD);
            ag0 = *(const float4*)(asrc + kc);
            ag1 = *(const float4*)(asrc + kc + 4);
        }
        // compute: 4 K-steps x 8 N-subtiles = 32 WMMAs from LDS
        #pragma unroll
        for (int k0 = 0; k0 < 16; k0 += 4) {
            const int kp0 = (k0 >> 1) + half;    // chunk-local K-pair for this lane-half
            const v2f a = *(const v2f*)&ldsA[buf][wave * 256 + kp0 * 32 + m_local * 2];
            #pragma unroll
            for (int j = 0; j < 8; ++j) {
                const v2f bf = *(const v2f*)&ldsW[buf][kp0 * 288 + (j * 16 + m_local) * 2];
                acc[j] = __builtin_amdgcn_wmma_f32_16x16x4_f32(
                    /*neg_a=*/false, a, /*neg_b=*/false, bf,
                    /*c_mod=*/(short)0, acc[j], /*reuse_a=*/false, /*reuse_b=*/false);
            }
        }
        // stage next chunk into the other buffer; single barrier per chunk
        if (c + 1 < NC) {
            const int nb = (c + 1) & 1;
            float* wd = &ldsW[nb][kpw * 288 + nqw * 2];
            float2 t;
            t.x = wg0.x; t.y = wg1.x; *(float2*)(wd + 0) = t;
            t.x = wg0.y; t.y = wg1.y; *(float2*)(wd + 2) = t;
            t.x = wg0.z; t.y = wg1.z; *(float2*)(wd + 4) = t;
            t.x = wg0.w; t.y = wg1.w; *(float2*)(wd + 6) = t;
            float* ad = &ldsA[nb][(ra >> 4) * 256 + (hca >> 1) * 32 + (ra & 15) * 2];
            t.x = ag0.x; t.y = ag0.y; *(float2*)(ad + 0 * 32) = t;
            t.x = ag0.z; t.y = ag0.w; *(float2*)(ad + 1 * 32) = t;
            t.x = ag1.x; t.y = ag1.y; *(float2*)(ad + 2 * 32) = t;
            t.x = ag1.z; t.y = ag1.w; *(float2*)(ad + 3 * 32) = t;
        }
        __syncthreads();
    }

    // epilogue: bias + validity mask (select, so poisoned padding rows -> 0)
    const int bidx = row_tile >> 7;              // batch of this 16-row tile
    const int nw   = nwords[bidx];
    #pragma unroll
    for (int j = 0; j < 8; ++j) {
        const int col = n0 + j * 16 + m_local;
        const float bv = bias[col];
        #pragma unroll
        for (int r = 0; r < 8; ++r) {
            const int m = row_tile * 16 + r + 8 * half;   // global row
            const bool valid = (m & (SS - 1)) < nw;
            out[(size_t)m * DD + col] = valid ? (acc[j][r] + bv) : 0.0f;
        }
    }
}

// ---------------------------------------------------------------------------
// Launch
// ---------------------------------------------------------------------------
extern "C" void kernel_launch(void* const* d_in, const int* in_sizes, int n_in,
                              void* d_out, int out_size, void* d_ws, size_t ws_size,
                              hipStream_t stream) {
    const float* frame_input = (const float*)d_in[0];   // [B,S,D] f32
    const float* w_score     = (const float*)d_in[1];   // [2D,1]  f32
    const float* b_score     = (const float*)d_in[2];   // [1]     f32
    const float* w_comb      = (const float*)d_in[3];   // [D,D]   f32
    const float* b_comb      = (const float*)d_in[4];   // [D]     f32

    float* out = (float*)d_out;                         // words [B,S,D] f32
    int* out_nwords = (int*)d_out + (size_t)BB * SS * DD;  // n_words [B] int32

    // workspace layout
    float* u        = (float*)d_ws;                     // B*S
    float* v        = u + BB * SS;                      // B*S
    int*   segstart = (int*)(v + BB * SS);              // B*S
    int*   nwords   = segstart + BB * SS;               // B
    size_t small_bytes = ((size_t)BB * SS * 3 + BB) * 4;
    small_bytes = (small_bytes + 255) & ~(size_t)255;
    float* seg_mean = (float*)((char*)d_ws + small_bytes);  // B*S*D (64 MB)

    // K1: 1 wave / frame
    k1_scores<<<dim3((BB * SS) / 8), dim3(256), 0, stream>>>(frame_input, w_score, u, v);

    // K2: 1 block / batch
    k2_scan<<<dim3(BB), dim3(256), 0, stream>>>(u, v, b_score, segstart, nwords, out_nwords);

    // K3: 1 wave / (batch, word slot)
    k3_segmean<<<dim3((BB * SS) / 8), dim3(256), 0, stream>>>(frame_input, segstart, nwords, seg_mean);

    // K4: WMMA GEMM, grid = (M/128, N/128)
    k4_gemm<<<dim3((BB * SS) / 128, DD / 128), dim3(256), 0, stream>>>(
        seg_mean, w_comb, b_comb, nwords, out);
}